// SOM_214748365211
// MI455X (gfx1250) — compile-verified
//
#include <hip/hip_runtime.h>
#include <hip/hip_bf16.h>
#include <stdint.h>

// SOM step: M=N=256, DIM=512. weights: (65536, 512) f32.
// out = [winner(512) | new_weights(65536*512)] f32.
#define SOM_GRID   256
#define SOM_DIMF   512                 // floats per row
#define SOM_ROWS   (SOM_GRID * SOM_GRID)

typedef float v2f __attribute__((ext_vector_type(2)));
typedef float v8f __attribute__((ext_vector_type(8)));

// ---------- CDNA5 async global->LDS helpers (inline asm, toolchain-portable) ----------

typedef const __attribute__((address_space(3))) void* lds_cptr_t;

static __device__ __forceinline__ uint32_t lds_off(const void* p) {
  // generic -> LDS addrspace cast, then ptrtoint: yields 32-bit DS byte offset
  return (uint32_t)(uintptr_t)(lds_cptr_t)p;
}

static __device__ __forceinline__ void async_copy_b128(const void* gsrc, void* ldst) {
  // GLOBAL_LOAD_ASYNC_TO_LDS_B128: per-lane 16B into per-lane LDS address (ASYNCcnt).
  asm volatile("global_load_async_to_lds_b128 %0, %1, off"
               :
               : "v"(lds_off(ldst)), "v"((unsigned long long)(uintptr_t)gsrc)
               : "memory");
}

static __device__ __forceinline__ void wait_async_le16() {
  asm volatile("s_wait_asynccnt 0x10" ::: "memory");
}
static __device__ __forceinline__ void wait_async_le0() {
  asm volatile("s_wait_asynccnt 0x0" ::: "memory");
}

static __device__ __forceinline__ unsigned long long shfl_xor_u64(unsigned long long v, int m) {
  unsigned lo = __shfl_xor((unsigned)v, m, 32);
  unsigned hi = __shfl_xor((unsigned)(v >> 32), m, 32);
  return ((unsigned long long)hi << 32) | lo;
}

// ---------- kernel 1: reset the packed (dist,idx) atomic-min cell ----------

__global__ void som_init_kernel(unsigned long long* res) {
  *res = ~0ull;
}

// ---------- kernel 2: BMU search (argmin of ||w_r - x||^2) ----------
// Each wave owns 16-row tiles. Lane L accumulates exact sum((w-x)^2) partials for
// row L%16 over column-half L/16. The 32 partial pairs form a 16x4 f32 A-matrix;
// one V_WMMA_F32_16X16X4_F32 against B=ones produces all 16 row sums at once.
// Tiles are async-DMA'd global->LDS in 8KiB double-buffered column chunks.

#define K2_BLOCKS   256
#define K2_WAVES    8
#define K2_TOTWAVES (K2_BLOCKS * K2_WAVES)      // 2048
#define TILES       (SOM_ROWS / 16)             // 4096
#define TILES_PW    (TILES / K2_TOTWAVES)       // 2 tiles per wave
#define CHUNK_COLS  128                         // floats per row per chunk
#define ROW_BYTES   (CHUNK_COLS * 4 + 16)       // 528: 132 dwords, kills 64-bank conflicts
#define CHUNK_BYTES (16 * ROW_BYTES)            // 8448

__global__ __launch_bounds__(256) void som_argmin_kernel(
    const float* __restrict__ x, const float* __restrict__ w,
    unsigned long long* __restrict__ res) {
  __shared__ __align__(16) unsigned char stage[K2_WAVES][2][CHUNK_BYTES];  // ~132 KiB
  __shared__ float xs[SOM_DIMF];
  __shared__ unsigned long long wres[K2_WAVES];

  const int wave  = threadIdx.x >> 5;
  const int lane  = threadIdx.x & 31;
  const int gwave = blockIdx.x * K2_WAVES + wave;

  if (threadIdx.x < SOM_DIMF / 4)
    ((float4*)xs)[threadIdx.x] = ((const float4*)x)[threadIdx.x];
  __syncthreads();

  const int row16 = lane & 15;                  // row within tile this lane scores
  const int half  = lane >> 4;                  // which 64-float4 column half

  auto prefetch = [&](int s, int buf) {
    const int tile = gwave + (s >> 2) * K2_TOTWAVES;
    const int c    = s & 3;
    const float* g = w + (size_t)tile * (16 * SOM_DIMF) + c * CHUNK_COLS + lane * 4;
    unsigned char* l = &stage[wave][buf][lane * 16];
#pragma unroll
    for (int j = 0; j < 16; ++j)                // row j of tile: coalesced 512B segment
      async_copy_b128(g + (size_t)j * SOM_DIMF, l + j * ROW_BYTES);
  };

  float bestD = 3.402823466e38f;
  unsigned bestI = 0u;
  float4 acc = {0.f, 0.f, 0.f, 0.f};

  const int NS = TILES_PW * 4;                  // 8 chunks per wave
  int buf = 0;
  prefetch(0, 0);
  for (int s = 0; s < NS; ++s) {
    if (s + 1 < NS) { prefetch(s + 1, buf ^ 1); wait_async_le16(); }
    else            { wait_async_le0(); }

    const int c = s & 3;
    const unsigned char* lb = &stage[wave][buf][row16 * ROW_BYTES + half * (64 * 4)];
    const float* xb = &xs[c * CHUNK_COLS + half * 64];
#pragma unroll
    for (int i = 0; i < 16; ++i) {
      const float4 wv = *(const float4*)(lb + i * 16);
      const float4 xv = *(const float4*)(xb + i * 4);
      const float dx = wv.x - xv.x, dy = wv.y - xv.y;
      const float dz = wv.z - xv.z, dw = wv.w - xv.w;
      acc.x = fmaf(dx, dx, acc.x); acc.y = fmaf(dy, dy, acc.y);
      acc.z = fmaf(dz, dz, acc.z); acc.w = fmaf(dw, dw, acc.w);
    }

    if (c == 3) {                               // tile complete: WMMA row-sum reduction
      v2f A; A.x = acc.x + acc.y; A.y = acc.z + acc.w;   // A[16x4] of partials
      v2f B; B.x = 1.0f; B.y = 1.0f;                     // B[4x16] = ones
      v8f C = {};
      // C[i][j] = sum_k A[i][k]: all 16 row distances in one matrix op
      C = __builtin_amdgcn_wmma_f32_16x16x4_f32(false, A, false, B,
                                                (short)0, C, false, false);
      const int tile  = gwave + (s >> 2) * K2_TOTWAVES;
      const int rbase = tile * 16 + half * 8;   // lane holds rows rbase..rbase+7 in C[k]
#pragma unroll
      for (int k = 0; k < 8; ++k) {
        const float sv = C[k];
        if (sv < bestD) { bestD = sv; bestI = (unsigned)(rbase + k); }
      }
      acc.x = acc.y = acc.z = acc.w = 0.f;
    }
    buf ^= 1;
  }

  // lane-local best -> wave -> block -> global packed u64 atomic min
  unsigned long long p = ((unsigned long long)__float_as_uint(bestD) << 32) | bestI;
#pragma unroll
  for (int off = 16; off >= 1; off >>= 1) {
    const unsigned long long q = shfl_xor_u64(p, off);
    p = (q < p) ? q : p;
  }
  if (lane == 0) wres[wave] = p;
  __syncthreads();
  if (threadIdx.x == 0) {
    unsigned long long m = wres[0];
#pragma unroll
    for (int i = 1; i < K2_WAVES; ++i) m = (wres[i] < m) ? wres[i] : m;
    atomicMin(res, m);                          // global_atomic_min_u64
  }
}

// ---------- kernel 3: weight update + winner copy ----------

__global__ __launch_bounds__(256) void som_update_kernel(
    const float* __restrict__ x, const float* __restrict__ w,
    const int* __restrict__ itp, const unsigned long long* __restrict__ res,
    float* __restrict__ out) {
  const unsigned bmu = (unsigned)(*res & 0xffffffffull);
  const float lr     = 1.0f - (float)(*itp) * (1.0f / 1000.0f);
  const float alpha  = 0.3f * lr;
  const float sigma  = 128.0f * lr;             // max(M,N)/2 * lr
  const float inv2   = 1.0f / (sigma * sigma);
  const int bi = (int)(bmu >> 8), bj = (int)(bmu & 255u);

  const float4* w4 = (const float4*)w;
  const float4* x4 = (const float4*)x;
  float4* o4 = (float4*)(out + SOM_DIMF);       // new_weights region (16B aligned)

  const int total  = SOM_ROWS * (SOM_DIMF / 4); // 8,388,608 float4s
  const int stride = gridDim.x * blockDim.x;
  for (int idx = blockIdx.x * blockDim.x + threadIdx.x; idx < total; idx += stride) {
    const int r = idx >> 7;
    const int c = idx & 127;
    const int di = (r >> 8) - bi;
    const int dj = (r & 255) - bj;
    const float f = alpha * __expf(-(float)(di * di + dj * dj) * inv2);
    const float4 wv = w4[idx];
    const float4 xv = x4[c];
    float4 o;
    o.x = fmaf(f, xv.x - wv.x, wv.x);
    o.y = fmaf(f, xv.y - wv.y, wv.y);
    o.z = fmaf(f, xv.z - wv.z, wv.z);
    o.w = fmaf(f, xv.w - wv.w, wv.w);
    o4[idx] = o;
  }

  // winner = pre-update weights[bmu] (inputs never mutated)
  if (blockIdx.x == 0 && threadIdx.x < SOM_DIMF / 4) {
    ((float4*)out)[threadIdx.x] = w4[(size_t)bmu * (SOM_DIMF / 4) + threadIdx.x];
  }
}

// ---------- launcher ----------

extern "C" void kernel_launch(void* const* d_in, const int* in_sizes, int n_in,
                              void* d_out, int out_size, void* d_ws, size_t ws_size,
                              hipStream_t stream) {
  const float* x = (const float*)d_in[0];
  const float* w = (const float*)d_in[1];
  // d_in[2] = y (unused by the math)
  const int* it = (const int*)d_in[3];
  float* out = (float*)d_out;
  unsigned long long* res = (unsigned long long*)d_ws;

  som_init_kernel<<<1, 1, 0, stream>>>(res);
  som_argmin_kernel<<<K2_BLOCKS, 256, 0, stream>>>(x, w, res);
  som_update_kernel<<<2048, 256, 0, stream>>>(x, w, it, res, out);
}